// CDFVarianceLoss_90503550861915
// MI455X (gfx1250) — compile-verified
//
#include <hip/hip_runtime.h>
#include <hip/hip_bf16.h>

#define BINS 64
#define NPB  65536          // elements per sample (256*256)
#define NSAMP 16
#define EPS 1e-6f
#define INV_SIGMA 20.0f     // 1/0.05
#define NEG_HALF_LOG2E (-0.7213475204444817f)  // -0.5 * log2(e)
#define INV_BINS_M1 (1.0f / 63.0f)

typedef __attribute__((ext_vector_type(2))) float v2f;
typedef __attribute__((ext_vector_type(8))) float v8f;

// ---------------------------------------------------------------------------
// Kernel 1: per-(tensor,sample) min/max reduction + zero the histogram region.
// grid = 32 blocks (t*16+s), block = 256 threads.
// ws layout: [0..63] minmax pairs (32 * {min,max}), [64..2111] h (32*64 floats)
// ---------------------------------------------------------------------------
__global__ void minmax_kernel(const float* __restrict__ pred,
                              const float* __restrict__ target,
                              float* __restrict__ ws) {
    __shared__ float smin[256];
    __shared__ float smax[256];

    const int pair = blockIdx.x;            // 0..31
    const int t = pair >> 4;                // 0=pred, 1=target
    const int s = pair & 15;
    const float* src = (t == 0 ? pred : target) + (size_t)s * NPB;

    const int tid = threadIdx.x;
    float mn = 3.4e38f, mx = -3.4e38f;
    for (int i = tid; i < NPB; i += 256) {
        float v = src[i];
        mn = fminf(mn, v);
        mx = fmaxf(mx, v);
    }
    smin[tid] = mn;
    smax[tid] = mx;
    __syncthreads();
    for (int off = 128; off > 0; off >>= 1) {
        if (tid < off) {
            smin[tid] = fminf(smin[tid], smin[tid + off]);
            smax[tid] = fmaxf(smax[tid], smax[tid + off]);
        }
        __syncthreads();
    }
    if (tid == 0) {
        ws[pair * 2 + 0] = smin[0];
        ws[pair * 2 + 1] = smax[0];
    }
    // zero this pair's 64 histogram bins (h starts at ws+64)
    if (tid < BINS) {
        ws[64 + pair * BINS + tid] = 0.0f;
    }
}

// ---------------------------------------------------------------------------
// Kernel 2: soft-histogram accumulation via V_WMMA_F32_16X16X4_F32.
//   D(16x16) = A(16x4) * ones(4x16) + C  ->  D[m][*] = sum_k A[m][k]
// Rows m of A = 16 bins (of one quarter), cols k = 4 consecutive elements.
// A VGPR layout (32-bit A 16x4, wave32): lanes 0-15 hold M=lane with
// {VGPR0=K0, VGPR1=K1}; lanes 16-31 hold M=lane-16 with {VGPR0=K2, VGPR1=K3}.
// C/D layout: VGPR r = row m=r (lanes 0-15) / m=r+8 (lanes 16-31), all
// columns identical (B==ones), so lane0/lane16 extract the bin sums.
// grid = 512 blocks (pair*16+chunk), block = 256 threads = 8 waves,
// wave handles 512 contiguous elements (4 per WMMA iteration).
// ---------------------------------------------------------------------------
__global__ void hist_kernel(const float* __restrict__ pred,
                            const float* __restrict__ target,
                            float* __restrict__ ws) {
    const int bid   = blockIdx.x;
    const int pair  = bid >> 4;             // 0..31
    const int chunk = bid & 15;             // 0..15 -> 4096 elements each
    const int t = pair >> 4;
    const int s = pair & 15;
    const float* src = (t == 0 ? pred : target) + (size_t)s * NPB;

    const float mn  = ws[pair * 2 + 0];
    const float mx  = ws[pair * 2 + 1];
    const float inv = 1.0f / (mx - mn + EPS);

    const int lane = threadIdx.x & 31;
    const int wave = threadIdx.x >> 5;
    const int binLocal = lane & 15;         // row m of A = bin (within quarter)
    const int kbase = (lane >> 4) * 2;      // K pair this lane supplies

    const int waveBase = chunk * 4096 + wave * 512;

    v8f acc0 = {}, acc1 = {}, acc2 = {}, acc3 = {};
    v2f ones; ones.x = 1.0f; ones.y = 1.0f;

    for (int e0 = waveBase; e0 < waveBase + 512; e0 += 4) {
        const float z0 = (src[e0 + kbase + 0] - mn) * inv;
        const float z1 = (src[e0 + kbase + 1] - mn) * inv;

        {   // bins 0..15
            const float c = (float)(binLocal + 0) * INV_BINS_M1;
            float d0 = (z0 - c) * INV_SIGMA, d1 = (z1 - c) * INV_SIGMA;
            v2f a; a.x = __builtin_amdgcn_exp2f(NEG_HALF_LOG2E * d0 * d0);
                   a.y = __builtin_amdgcn_exp2f(NEG_HALF_LOG2E * d1 * d1);
            acc0 = __builtin_amdgcn_wmma_f32_16x16x4_f32(false, a, false, ones,
                                                         (short)0, acc0, false, false);
        }
        {   // bins 16..31
            const float c = (float)(binLocal + 16) * INV_BINS_M1;
            float d0 = (z0 - c) * INV_SIGMA, d1 = (z1 - c) * INV_SIGMA;
            v2f a; a.x = __builtin_amdgcn_exp2f(NEG_HALF_LOG2E * d0 * d0);
                   a.y = __builtin_amdgcn_exp2f(NEG_HALF_LOG2E * d1 * d1);
            acc1 = __builtin_amdgcn_wmma_f32_16x16x4_f32(false, a, false, ones,
                                                         (short)0, acc1, false, false);
        }
        {   // bins 32..47
            const float c = (float)(binLocal + 32) * INV_BINS_M1;
            float d0 = (z0 - c) * INV_SIGMA, d1 = (z1 - c) * INV_SIGMA;
            v2f a; a.x = __builtin_amdgcn_exp2f(NEG_HALF_LOG2E * d0 * d0);
                   a.y = __builtin_amdgcn_exp2f(NEG_HALF_LOG2E * d1 * d1);
            acc2 = __builtin_amdgcn_wmma_f32_16x16x4_f32(false, a, false, ones,
                                                         (short)0, acc2, false, false);
        }
        {   // bins 48..63
            const float c = (float)(binLocal + 48) * INV_BINS_M1;
            float d0 = (z0 - c) * INV_SIGMA, d1 = (z1 - c) * INV_SIGMA;
            v2f a; a.x = __builtin_amdgcn_exp2f(NEG_HALF_LOG2E * d0 * d0);
                   a.y = __builtin_amdgcn_exp2f(NEG_HALF_LOG2E * d1 * d1);
            acc3 = __builtin_amdgcn_wmma_f32_16x16x4_f32(false, a, false, ones,
                                                         (short)0, acc3, false, false);
        }
    }

    // Extract bin sums: lane 0 holds rows 0..7 (VGPR r -> bin r),
    // lane 16 holds rows 8..15 (VGPR r -> bin r+8). All columns identical.
    float* h = ws + 64 + pair * BINS;
    if (lane == 0 || lane == 16) {
        const int rowOff = (lane == 16) ? 8 : 0;
        #pragma unroll
        for (int r = 0; r < 8; ++r) {
            atomicAdd(&h[ 0 + rowOff + r], acc0[r]);
            atomicAdd(&h[16 + rowOff + r], acc1[r]);
            atomicAdd(&h[32 + rowOff + r], acc2[r]);
            atomicAdd(&h[48 + rowOff + r], acc3[r]);
        }
    }
}

// ---------------------------------------------------------------------------
// Kernel 3: normalize + cumsum + mean squared CDF difference. 1 block.
// ---------------------------------------------------------------------------
__global__ void finalize_kernel(const float* __restrict__ ws,
                                float* __restrict__ out) {
    __shared__ float partial[16];
    const int tid = threadIdx.x;
    if (tid < NSAMP) {
        const float* hx = ws + 64 + tid * BINS;           // pred pair = s
        const float* hy = ws + 64 + (16 + tid) * BINS;    // target pair = 16+s
        float sx = 0.0f, sy = 0.0f;
        for (int j = 0; j < BINS; ++j) {
            sx += hx[j] + EPS;
            sy += hy[j] + EPS;
        }
        const float isx = 1.0f / sx, isy = 1.0f / sy;
        float cx = 0.0f, cy = 0.0f, acc = 0.0f;
        for (int j = 0; j < BINS; ++j) {
            cx += (hx[j] + EPS) * isx;
            cy += (hy[j] + EPS) * isy;
            float d = cx - cy;
            acc += d * d;
        }
        partial[tid] = acc;
    }
    __syncthreads();
    if (tid == 0) {
        float total = 0.0f;
        for (int i = 0; i < NSAMP; ++i) total += partial[i];
        out[0] = total / (float)(NSAMP * BINS);
    }
}

extern "C" void kernel_launch(void* const* d_in, const int* in_sizes, int n_in,
                              void* d_out, int out_size, void* d_ws, size_t ws_size,
                              hipStream_t stream) {
    const float* pred   = (const float*)d_in[0];
    const float* target = (const float*)d_in[1];
    float* ws  = (float*)d_ws;
    float* out = (float*)d_out;

    minmax_kernel<<<32, 256, 0, stream>>>(pred, target, ws);
    hist_kernel<<<512, 256, 0, stream>>>(pred, target, ws);
    finalize_kernel<<<1, 32, 0, stream>>>(ws, out);
}